// Canny_66649302499584
// MI455X (gfx1250) — compile-verified
//
#include <hip/hip_runtime.h>
#include <math.h>

#define N_IMG 16
#define H 512
#define W 512
#define HW (H * W)
#define NHW (N_IMG * HW)
#define WPR (W / 32)          /* 32-bit words per row   = 16   */
#define WPI (H * WPR)         /* words per image        = 8192 */

typedef float v2f __attribute__((ext_vector_type(2)));
typedef float v8f __attribute__((ext_vector_type(8)));

// ---------------------------------------------------------------------------
// 1. Grayscale: 0.299 R + 0.587 G + 0.114 B
// ---------------------------------------------------------------------------
__global__ __launch_bounds__(256) void gray_kernel(const float* __restrict__ x,
                                                   float* __restrict__ gray) {
  int p = blockIdx.x * 256 + threadIdx.x;
  if (p >= NHW) return;
  int n = p >> 18;                 // HW == 2^18
  int rem = p & (HW - 1);
  const float* base = x + (size_t)n * 3 * HW + rem;
  gray[p] = 0.299f * base[0] + 0.587f * base[HW] + 0.114f * base[2 * HW];
}

// ---------------------------------------------------------------------------
// 2. 5x5 Gaussian blur (reflect pad), separable.
//    Horizontal pass: VALU into LDS.  Vertical pass: out = Wv(16x20)*T(20x16)
//    as 5 chained V_WMMA_F32_16X16X4_F32.  One wave per 16x16 tile.
// ---------------------------------------------------------------------------
__device__ __forceinline__ float gw_sel(int d, float g0, float g1, float g2) {
  // g[d] for d in [0,4] where g = {g0,g1,g2,g1,g0}; 0 otherwise
  float w = 0.0f;
  w = (d == 0 || d == 4) ? g0 : w;
  w = (d == 1 || d == 3) ? g1 : w;
  w = (d == 2) ? g2 : w;
  return w;
}

__global__ __launch_bounds__(32) void blur_wmma_kernel(const float* __restrict__ gray,
                                                       float* __restrict__ blur,
                                                       float g0, float g1, float g2) {
  __shared__ float S[20][20];      // input tile (with 2-halo, reflect indexed)
  __shared__ float T[20][16];      // after horizontal pass
  int blk = blockIdx.x;            // 16 images * 32*32 tiles
  int n = blk >> 10;
  int t = blk & 1023;
  int ty = (t >> 5) << 4;
  int tx = (t & 31) << 4;
  const float* img = gray + (size_t)n * HW;
  int lane = threadIdx.x;

  for (int idx = lane; idx < 400; idx += 32) {
    int r = idx / 20, c = idx - r * 20;
    int yy = ty + r - 2; yy = (yy < 0) ? -yy : ((yy >= H) ? (2 * H - 2 - yy) : yy);
    int xx = tx + c - 2; xx = (xx < 0) ? -xx : ((xx >= W) ? (2 * W - 2 - xx) : xx);
    S[r][c] = img[yy * W + xx];
  }
  __syncthreads();

  for (int idx = lane; idx < 320; idx += 32) {
    int r = idx >> 4, c = idx & 15;
    T[r][c] = g0 * S[r][c] + g1 * S[r][c + 1] + g2 * S[r][c + 2] +
              g1 * S[r][c + 3] + g0 * S[r][c + 4];
  }
  __syncthreads();

  // Vertical pass via WMMA.  A-layout (16x4 f32): lane l -> M = l%16,
  // K = {0,1} for lanes 0-15, {2,3} for lanes 16-31 (ISA 7.12.2).
  int m = lane & 15;
  int kb = (lane < 16) ? 0 : 2;
  v8f acc = {0.f, 0.f, 0.f, 0.f, 0.f, 0.f, 0.f, 0.f};
#pragma unroll
  for (int j = 0; j < 5; ++j) {
    v2f a, b;
    a.x = gw_sel(4 * j + kb - m,     g0, g1, g2);   // Wv[m][4j+kb]
    a.y = gw_sel(4 * j + kb + 1 - m, g0, g1, g2);   // Wv[m][4j+kb+1]
    b.x = T[4 * j + kb][m];                         // B rows 4j+{0|2}
    b.y = T[4 * j + kb + 1][m];                     // B rows 4j+{1|3}
    acc = __builtin_amdgcn_wmma_f32_16x16x4_f32(
        /*neg_a=*/false, a, /*neg_b=*/false, b,
        /*c_mod=*/(short)0, acc, /*reuse_a=*/false, /*reuse_b=*/false);
  }

  // D layout: VGPR v -> row v (lanes 0-15) / row v+8 (lanes 16-31), col = lane%16
  float* obase = blur + (size_t)n * HW;
  int rb = (lane < 16) ? 0 : 8;
#pragma unroll
  for (int v = 0; v < 8; ++v) {
    obase[(ty + rb + v) * W + tx + m] = acc[v];
  }
}

// ---------------------------------------------------------------------------
// 3. Sobel (edge pad) -> magnitude + quantized direction index
// ---------------------------------------------------------------------------
__global__ __launch_bounds__(256) void sobel_kernel(const float* __restrict__ blur,
                                                    float* __restrict__ mag,
                                                    unsigned char* __restrict__ dir) {
  int p = blockIdx.x * 256 + threadIdx.x;
  if (p >= NHW) return;
  int n = p >> 18;
  int rem = p & (HW - 1);
  int y = rem >> 9, x = rem & (W - 1);
  const float* img = blur + (size_t)n * HW;
  int ym = (y > 0) ? y - 1 : 0, yp = (y < H - 1) ? y + 1 : H - 1;
  int xm = (x > 0) ? x - 1 : 0, xp = (x < W - 1) ? x + 1 : W - 1;
  float a00 = img[ym * W + xm], a01 = img[ym * W + x], a02 = img[ym * W + xp];
  float a10 = img[y * W + xm],                          a12 = img[y * W + xp];
  float a20 = img[yp * W + xm], a21 = img[yp * W + x], a22 = img[yp * W + xp];
  float gx = (a02 - a00) + 2.f * (a12 - a10) + (a22 - a20);
  float gy = (a20 - a00) + 2.f * (a21 - a01) + (a22 - a02);
  mag[p] = sqrtf(gx * gx + gy * gy + 1e-6f);
  float ang = rintf(atan2f(gy, gx) * (float)(4.0 / M_PI));  // round half-even
  int ai = (int)ang;                                        // in [-4, 4]
  dir[p] = (unsigned char)((ai + 8) & 7);                   // mod 8
}

// ---------------------------------------------------------------------------
// 4. NMS (zero-pad neighbors) -> final magnitude output
// ---------------------------------------------------------------------------
__constant__ int c_dy[8] = {0, -1, -1, -1, 0, 1, 1, 1};
__constant__ int c_dx[8] = {1,  1,  0, -1, -1, -1, 0, 1};

__global__ __launch_bounds__(256) void nms_kernel(const float* __restrict__ mag,
                                                  const unsigned char* __restrict__ dir,
                                                  float* __restrict__ outmag) {
  int p = blockIdx.x * 256 + threadIdx.x;
  if (p >= NHW) return;
  int n = p >> 18;
  int rem = p & (HW - 1);
  int y = rem >> 9, x = rem & (W - 1);
  const float* im = mag + (size_t)n * HW;
  float m = im[rem];
  int pos = dir[p];
  int neg = (pos + 4) & 7;
  int y1 = y + c_dy[pos], x1 = x + c_dx[pos];
  int y2 = y + c_dy[neg], x2 = x + c_dx[neg];
  float mp = (y1 >= 0 && y1 < H && x1 >= 0 && x1 < W) ? im[y1 * W + x1] : 0.f;
  float mn = (y2 >= 0 && y2 < H && x2 >= 0 && x2 < W) ? im[y2 * W + x2] : 0.f;
  float keep = (fminf(m - mp, m - mn) > 0.f) ? 1.f : 0.f;
  outmag[p] = m * keep;
}

// ---------------------------------------------------------------------------
// 5. Pack double-threshold masks to 1 bit/pixel
// ---------------------------------------------------------------------------
__global__ __launch_bounds__(256) void pack_kernel(const float* __restrict__ outmag,
                                                   unsigned* __restrict__ sbits,
                                                   unsigned* __restrict__ wbits) {
  int wdx = blockIdx.x * 256 + threadIdx.x;
  if (wdx >= NHW / 32) return;
  const float* base = outmag + (size_t)wdx * 32;
  unsigned s = 0, w = 0;
#pragma unroll
  for (int k = 0; k < 32; ++k) {
    float m = base[k];
    unsigned hi = (m > 0.2f) ? 1u : 0u;
    unsigned lo = (m > 0.1f) ? 1u : 0u;
    s |= hi << k;
    w |= ((lo & ~hi) & 1u) << k;      // weak-only (edges == 0.5)
  }
  sbits[wdx] = s;
  wbits[wdx] = w;
}

// ---------------------------------------------------------------------------
// 6. Hysteresis: whole image bit-packed in LDS, up to 256 dilation steps,
//    deterministic early exit at fixed point.  One workgroup per image.
//    S_{t+1} = S_t | (weak & dil3x3(S_t))  (center incl. is harmless:
//    weak ∧ strong = ∅), identical to the reference's hm fixed point.
// ---------------------------------------------------------------------------
__device__ __forceinline__ unsigned hrow(const unsigned* S, int y, int i) {
  const unsigned* row = S + (y << 4);
  unsigned a = row[i];
  unsigned l = (i > 0) ? row[i - 1] : 0u;
  unsigned r = (i < WPR - 1) ? row[i + 1] : 0u;
  return a | (a << 1) | (l >> 31) | (a >> 1) | (r << 31);
}

__global__ __launch_bounds__(256) void hyst_kernel(unsigned* sbits,
                                                   const unsigned* __restrict__ wbits) {
  extern __shared__ unsigned sm[];
  unsigned* S = sm;                // 8192 words = 32 KB (strong bits)
  unsigned* Wk = sm + WPI;         // 8192 words (weak-only bits)
  unsigned* flag = sm + 2 * WPI;
  int n = blockIdx.x;
  unsigned* gs = sbits + (size_t)n * WPI;
  const unsigned* gw = wbits + (size_t)n * WPI;
  int t = threadIdx.x;
  for (int k = t; k < WPI; k += 256) { S[k] = gs[k]; Wk[k] = gw[k]; }
  if (t == 0) *flag = 0u;
  __syncthreads();

  int base = t * 32;               // each thread owns 32 words (2 rows)
  for (int it = 0; it < 256; ++it) {
    unsigned nw[32];
    unsigned changed = 0u;
#pragma unroll
    for (int k = 0; k < 32; ++k) {
      int w = base + k;
      int y = w >> 4, i = w & (WPR - 1);
      unsigned d = hrow(S, y, i);
      if (y > 0)     d |= hrow(S, y - 1, i);
      if (y < H - 1) d |= hrow(S, y + 1, i);
      unsigned cur = S[w];
      unsigned nv = cur | (Wk[w] & d);
      nw[k] = nv;
      changed |= (nv ^ cur);
    }
    __syncthreads();               // all reads of S complete
#pragma unroll
    for (int k = 0; k < 32; ++k) S[base + k] = nw[k];
    if (changed) atomicOr(flag, 1u);
    __syncthreads();               // writes + flag visible
    unsigned c = *flag;
    __syncthreads();               // everyone has read the flag
    if (t == 0) *flag = 0u;
    if (!c) break;                 // uniform, data-dependent only
  }
  __syncthreads();
  for (int k = t; k < WPI; k += 256) gs[k] = S[k];
}

// ---------------------------------------------------------------------------
// 7. Unpack final strong bits -> float edges output
// ---------------------------------------------------------------------------
__global__ __launch_bounds__(256) void unpack_kernel(const unsigned* __restrict__ sbits,
                                                     float* __restrict__ edges) {
  int p = blockIdx.x * 256 + threadIdx.x;
  if (p >= NHW) return;
  unsigned w = sbits[p >> 5];
  edges[p] = ((w >> (p & 31)) & 1u) ? 1.0f : 0.0f;
}

// ---------------------------------------------------------------------------
// Launch
// ---------------------------------------------------------------------------
extern "C" void kernel_launch(void* const* d_in, const int* in_sizes, int n_in,
                              void* d_out, int out_size, void* d_ws, size_t ws_size,
                              hipStream_t stream) {
  (void)in_sizes; (void)n_in; (void)out_size; (void)ws_size;
  const float* x = (const float*)d_in[0];
  float* outmag  = (float*)d_out;          // magnitude: NHW floats
  float* outedge = outmag + NHW;           // edges:     NHW floats

  char* ws = (char*)d_ws;
  float* bufA = (float*)ws;                                 // gray, then mag
  float* bufB = (float*)(ws + (size_t)NHW * 4);             // blurred
  unsigned char* dir = (unsigned char*)(ws + (size_t)NHW * 8);
  unsigned* sbits = (unsigned*)(ws + (size_t)NHW * 8 + NHW);
  unsigned* wbits = sbits + (NHW / 32);
  // total workspace: 16+16+4+0.5+0.5 = 37 MB

  // Gaussian weights, computed in f32 exactly as the reference does.
  float e0 = expf(-2.0f), e1 = expf(-0.5f);
  float s = 2.f * e0 + 2.f * e1 + 1.f;
  float g0 = e0 / s, g1 = e1 / s, g2 = 1.f / s;

  gray_kernel<<<NHW / 256, 256, 0, stream>>>(x, bufA);
  blur_wmma_kernel<<<N_IMG * 32 * 32, 32, 0, stream>>>(bufA, bufB, g0, g1, g2);
  sobel_kernel<<<NHW / 256, 256, 0, stream>>>(bufB, bufA, dir);
  nms_kernel<<<NHW / 256, 256, 0, stream>>>(bufA, dir, outmag);
  pack_kernel<<<(NHW / 32) / 256, 256, 0, stream>>>(outmag, sbits, wbits);
  hyst_kernel<<<N_IMG, 256, (2 * WPI + 1) * sizeof(unsigned), stream>>>(sbits, wbits);
  unpack_kernel<<<NHW / 256, 256, 0, stream>>>(sbits, outedge);
}